// OurQuantizer_35424890257551
// MI455X (gfx1250) — compile-verified
//
#include <hip/hip_runtime.h>
#include <stdint.h>

// ---------------------------------------------------------------------------
// Rowwise int4 symmetric quantization (QuaRot sym_quant_half semantics).
//   x: [N, 8192] fp16 ; out: q int8 [N*8192] followed by fp16 scales [N]
// Memory-bound: ~402 MB total traffic -> ~17us @ 23.3 TB/s.
// CDNA5 path: async global->LDS double-buffered row pipeline (ASYNCcnt),
// packed fp16 abs-max (v_pk_max_num_f16), wave32 shfl_xor reduction,
// non-temporal output stores (write-once stream, keep L2 for input).
// ---------------------------------------------------------------------------

#define KCOLS 8192
#define TPB 256
#define CHUNKS_PER_ROW (KCOLS * 2 / 16)          // 1024 x 16B chunks per row
#define CHUNKS_PER_THREAD (CHUNKS_PER_ROW / TPB) // 4
#define NBLOCKS 2048

#if defined(__has_builtin)
#if __has_builtin(__builtin_amdgcn_global_load_async_to_lds_b128) && \
    __has_builtin(__builtin_amdgcn_s_wait_asynccnt)
#define HAVE_ASYNC_LDS 1
#endif
#endif

// Builtin expects pointers to gcc-style vector-of-4-int, AS1 (global) / AS3 (LDS).
typedef int v4i __attribute__((vector_size(16)));
typedef __attribute__((address_space(1))) v4i gv4i;
typedef __attribute__((address_space(3))) v4i lv4i;

// Plain vector types for nontemporal stores / packed halves.
typedef unsigned v2u __attribute__((vector_size(8)));
typedef _Float16 h2 __attribute__((ext_vector_type(2)));

__device__ __forceinline__ unsigned pk_abs_max(unsigned m, unsigned v) {
  v &= 0x7FFF7FFFu; // fp16 abs = clear sign bits (packed)
  h2 r = __builtin_elementwise_max(__builtin_bit_cast(h2, m),
                                   __builtin_bit_cast(h2, v)); // v_pk_max_num_f16
  return __builtin_bit_cast(unsigned, r);
}

__device__ __forceinline__ unsigned quant_pack4(unsigned w0, unsigned w1, float inv) {
  h2 a = __builtin_bit_cast(h2, w0);
  h2 b = __builtin_bit_cast(h2, w1);
  float f0 = (float)a.x, f1 = (float)a.y;
  float f2 = (float)b.x, f3 = (float)b.y;
  // round-half-even (v_rndne), clamp folds to v_med3_f32
  int b0 = (int)fminf(fmaxf(rintf(f0 * inv), -8.0f), 7.0f);
  int b1 = (int)fminf(fmaxf(rintf(f1 * inv), -8.0f), 7.0f);
  int b2 = (int)fminf(fmaxf(rintf(f2 * inv), -8.0f), 7.0f);
  int b3 = (int)fminf(fmaxf(rintf(f3 * inv), -8.0f), 7.0f);
  return (b0 & 255) | ((b1 & 255) << 8) | ((b2 & 255) << 16) | ((b3 & 255) << 24);
}

// Reduce + quantize one row held in registers (d[0..3] = 32 halves/thread).
__device__ __forceinline__ void quantize_row(const uint4* d, int r, int tid,
                                             int lane, int wave, float* sred,
                                             int8_t* __restrict__ qout,
                                             _Float16* __restrict__ scales) {
  unsigned m = 0u;
#pragma unroll
  for (int i = 0; i < CHUNKS_PER_THREAD; ++i) {
    m = pk_abs_max(m, d[i].x);
    m = pk_abs_max(m, d[i].y);
    m = pk_abs_max(m, d[i].z);
    m = pk_abs_max(m, d[i].w);
  }
  h2 hm = __builtin_bit_cast(h2, m);
  float fm = fmaxf((float)hm.x, (float)hm.y);
#pragma unroll
  for (int off = 16; off >= 1; off >>= 1) // wave32 butterfly
    fm = fmaxf(fm, __shfl_xor(fm, off, 32));
  if (lane == 0) sred[wave] = fm;
  __syncthreads();
  float rowmax = sred[0];
#pragma unroll
  for (int w = 1; w < TPB / 32; ++w) rowmax = fmaxf(rowmax, sred[w]);
  _Float16 sh = (_Float16)(rowmax * (1.0f / 7.0f)); // fp16 scale, RTE
  float inv = 1.0f / (float)sh;
  if (tid == 0) scales[r] = sh;

  v2u* qbase = (v2u*)(qout + (size_t)r * KCOLS);
#pragma unroll
  for (int i = 0; i < CHUNKS_PER_THREAD; ++i) {
    int c = tid + i * TPB;
    v2u outv = {quant_pack4(d[i].x, d[i].y, inv),
                quant_pack4(d[i].z, d[i].w, inv)};
    // write-once stream: bypass L2 retention (global_store_b64 ... nt)
    __builtin_nontemporal_store(outv, qbase + c);
  }
}

__global__ __launch_bounds__(TPB) void quarot_int4_rowquant_kernel(
    const uint4* __restrict__ x, int8_t* __restrict__ qout,
    _Float16* __restrict__ scales, int nrows) {
  const int tid = threadIdx.x;
  const int lane = tid & 31;
  const int wave = tid >> 5;
  __shared__ float sred[TPB / 32];

#if HAVE_ASYNC_LDS
  __shared__ uint4 sbuf[2][CHUNKS_PER_ROW]; // 2 x 16KB double buffer
  {
    int r0 = blockIdx.x;
    if (r0 < nrows) {
      const uint4* src = x + (size_t)r0 * CHUNKS_PER_ROW;
#pragma unroll
      for (int i = 0; i < CHUNKS_PER_THREAD; ++i) {
        int c = tid + i * TPB;
        __builtin_amdgcn_global_load_async_to_lds_b128(
            (gv4i*)(void*)(src + c), (lv4i*)(void*)&sbuf[0][c], 0, 0);
      }
    }
  }
  int it = 0;
  for (int r = blockIdx.x; r < nrows; r += gridDim.x, ++it) {
    int rn = r + (int)gridDim.x;
    if (rn < nrows) { // prefetch next row into the other buffer
      const uint4* src = x + (size_t)rn * CHUNKS_PER_ROW;
#pragma unroll
      for (int i = 0; i < CHUNKS_PER_THREAD; ++i) {
        int c = tid + i * TPB;
        __builtin_amdgcn_global_load_async_to_lds_b128(
            (gv4i*)(void*)(src + c), (lv4i*)(void*)&sbuf[(it + 1) & 1][c], 0, 0);
      }
      // async loads retire in order: <=4 pending => row r fully in LDS
      __builtin_amdgcn_s_wait_asynccnt(CHUNKS_PER_THREAD);
    } else {
      __builtin_amdgcn_s_wait_asynccnt(0);
    }
    __syncthreads(); // row r visible to all waves; prev-iter sred reads done
    uint4 d[CHUNKS_PER_THREAD];
#pragma unroll
    for (int i = 0; i < CHUNKS_PER_THREAD; ++i) // ds_load_b128 -> registers
      d[i] = sbuf[it & 1][tid + i * TPB];
    quantize_row(d, r, tid, lane, wave, sred, qout, scales);
  }
#else
  // Fallback: direct coalesced global loads, register-resident single pass.
  for (int r = blockIdx.x; r < nrows; r += gridDim.x) {
    const uint4* src = x + (size_t)r * CHUNKS_PER_ROW;
    uint4 d[CHUNKS_PER_THREAD];
#pragma unroll
    for (int i = 0; i < CHUNKS_PER_THREAD; ++i)
      d[i] = src[tid + i * TPB];
    __syncthreads(); // protect sred reuse across iterations
    quantize_row(d, r, tid, lane, wave, sred, qout, scales);
  }
#endif
}

extern "C" void kernel_launch(void* const* d_in, const int* in_sizes, int n_in,
                              void* d_out, int out_size, void* d_ws, size_t ws_size,
                              hipStream_t stream) {
  (void)n_in; (void)out_size; (void)d_ws; (void)ws_size;
  const uint4* x = (const uint4*)d_in[0];
  const int nrows = (int)((size_t)in_sizes[0] / KCOLS); // 16384
  int8_t* q = (int8_t*)d_out;                           // [nrows*8192] int8
  _Float16* scales = (_Float16*)((char*)d_out + (size_t)nrows * KCOLS); // [nrows] fp16
  int blocks = nrows < NBLOCKS ? nrows : NBLOCKS;
  quarot_int4_rowquant_kernel<<<blocks, TPB, 0, stream>>>(x, q, scales, nrows);
}